// TopologicalAttention_11063835755029
// MI455X (gfx1250) — compile-verified
//
#include <hip/hip_runtime.h>
#include <math.h>

// ---------------------------------------------------------------------------
// TopologicalAttention for MI455X (gfx1250, wave32, WMMA)
// fp32 matrix path via V_WMMA_F32_16X16X4_F32 (exact precision, matrix pipes)
// ---------------------------------------------------------------------------

typedef float v2f __attribute__((ext_vector_type(2)));
typedef float v8f __attribute__((ext_vector_type(8)));

#define B_    16
#define S_    64
#define H_    8
#define D_    64
#define ED_   512
#define NE_   2016          // S*(S-1)/2 edges
#define NCYC_ 1953          // NE_ - (S_-1)
#define ROWS_ (B_*S_)       // 1024

static __device__ __forceinline__ v8f wmma4(v2f a, v2f b, v8f c) {
  // D = A(16x4 f32) * B(4x16 f32) + C(16x16 f32)
  return __builtin_amdgcn_wmma_f32_16x16x4_f32(false, a, false, b, (short)0, c,
                                               false, false);
}

// ---------------------------------------------------------------------------
// Generic GEMM: out = A(MxK) @ W(KxN) + bias.  One wave per 16x16 tile.
// mode 0: out[m*N+n]   (row-major)
// mode 1: out in (B,H,S,D) layout for Q/K/V (m = b*S+s, n = h*D+d)
// ---------------------------------------------------------------------------
__global__ __launch_bounds__(32)
void gemm_wmma(const float* __restrict__ A, const float* __restrict__ W,
               const float* __restrict__ bias, float* __restrict__ out,
               int Kd, int N, int mode) {
  const int lane = threadIdx.x & 31;
  const int lr   = lane & 15;
  const int half = lane >> 4;
  const int n0 = blockIdx.x * 16;
  const int m0 = blockIdx.y * 16;
  v8f acc = {};
  for (int k = 0; k < Kd; k += 4) {
    const int ka = k + 2 * half;
    v2f a = *reinterpret_cast<const v2f*>(A + (m0 + lr) * Kd + ka);
    v2f b;
    b.x = W[ka * N + n0 + lr];
    b.y = W[(ka + 1) * N + n0 + lr];
    acc = wmma4(a, b, acc);
  }
  const int n  = n0 + lr;
  const float bv = bias[n];
#pragma unroll
  for (int j = 0; j < 8; ++j) {
    const int m = m0 + j + 8 * half;
    const float val = acc[j] + bv;
    if (mode == 0) {
      out[m * N + n] = val;
    } else {
      const int bb = m >> 6, s = m & 63, h = n >> 6, d = n & 63;
      out[(((bb * H_ + h) * S_ + s) * D_) + d] = val;
    }
  }
}

// ---------------------------------------------------------------------------
// Attention for one (b,h): scores = QK^T/8, softmax, attn = P@V.
// 4 waves; wave w owns query rows 16w..16w+15. Scores staged in LDS.
// Output written in (B,S,H*D) layout (the post-transpose reshape).
// ---------------------------------------------------------------------------
__global__ __launch_bounds__(128)
void attn_kernel(const float* __restrict__ Q, const float* __restrict__ K,
                 const float* __restrict__ V, float* __restrict__ attn_out) {
  __shared__ float sc[64][65];
  const int bh = blockIdx.x;
  const int b = bh / H_, h = bh % H_;
  const float* Qb = Q + bh * S_ * D_;
  const float* Kb = K + bh * S_ * D_;
  const float* Vb = V + bh * S_ * D_;
  const int tid  = threadIdx.x;
  const int wave = tid >> 5;
  const int lane = tid & 31;
  const int lr = lane & 15, half = lane >> 4;
  const int m0 = wave * 16;

  // ---- scores = Q @ K^T * (1/8) ----
  for (int nt = 0; nt < 4; ++nt) {
    const int n0 = nt * 16;
    v8f acc = {};
    for (int k = 0; k < D_; k += 4) {
      const int ka = k + 2 * half;
      v2f a = *reinterpret_cast<const v2f*>(Qb + (m0 + lr) * D_ + ka);
      v2f bf;                      // B[kk][n] = K[n][kk]
      bf.x = Kb[(n0 + lr) * D_ + ka];
      bf.y = Kb[(n0 + lr) * D_ + ka + 1];
      acc = wmma4(a, bf, acc);
    }
#pragma unroll
    for (int j = 0; j < 8; ++j)
      sc[m0 + j + 8 * half][n0 + lr] = acc[j] * 0.125f;
  }
  __syncthreads();

  // ---- softmax per row ----
  if (tid < 64) {
    float mx = -1e30f;
    for (int c = 0; c < 64; ++c) mx = fmaxf(mx, sc[tid][c]);
    float sum = 0.f;
    for (int c = 0; c < 64; ++c) {
      const float e = __expf(sc[tid][c] - mx);
      sc[tid][c] = e;
      sum += e;
    }
    const float inv = 1.f / sum;
    for (int c = 0; c < 64; ++c) sc[tid][c] *= inv;
  }
  __syncthreads();

  // ---- attn = P @ V ----
  for (int nt = 0; nt < 4; ++nt) {
    const int n0 = nt * 16;
    v8f acc = {};
    for (int k = 0; k < S_; k += 4) {
      const int ka = k + 2 * half;
      v2f a;
      a.x = sc[m0 + lr][ka];
      a.y = sc[m0 + lr][ka + 1];
      v2f bf;
      bf.x = Vb[ka * D_ + n0 + lr];
      bf.y = Vb[(ka + 1) * D_ + n0 + lr];
      acc = wmma4(a, bf, acc);
    }
#pragma unroll
    for (int j = 0; j < 8; ++j) {
      const int s = m0 + j + 8 * half;
      const int d = n0 + lr;
      attn_out[(b * S_ + s) * ED_ + h * D_ + d] = acc[j];
    }
  }
}

// ---------------------------------------------------------------------------
// rel[0,0] only (dead-code elimination: output depends on rel just through
// the violation scalar, which uses rel[0,0]).  One wave per 16 edges.
// pair = [Q00[EI], K00[EJ]] (128) -> relu(@Wr1+br1) (64) -> @Wr2+br2 (64)
// ---------------------------------------------------------------------------
__global__ __launch_bounds__(32)
void rel_kernel(const float* __restrict__ Q00, const float* __restrict__ K00,
                const float* __restrict__ Wr1, const float* __restrict__ br1,
                const float* __restrict__ Wr2, const float* __restrict__ br2,
                float* __restrict__ rel, float* __restrict__ xv) {
  __shared__ float hid[16][72];
  const int lane = threadIdx.x & 31;
  const int lr = lane & 15, half = lane >> 4;
  const int m0 = blockIdx.x * 16;

  // decode triangular edge index -> (i,j), i<j (np.triu_indices order)
  int i = 0, rem = m0 + lr;
  while (rem >= 63 - i) { rem -= 63 - i; ++i; }
  const int jn = i + 1 + rem;
  const float* qrow = Q00 + i * D_;    // (b=0,h=0) base
  const float* krow = K00 + jn * D_;

  // hidden = relu(pair @ Wr1 + br1)
  for (int nt = 0; nt < 4; ++nt) {
    const int n0 = nt * 16;
    v8f acc = {};
    for (int k = 0; k < 2 * D_; k += 4) {
      const int ka = k + 2 * half;           // even; {ka,ka+1} never straddle 64
      v2f a;
      if (ka < D_) { a.x = qrow[ka];       a.y = qrow[ka + 1]; }
      else         { a.x = krow[ka - D_];  a.y = krow[ka + 1 - D_]; }
      v2f bf;
      bf.x = Wr1[ka * D_ + n0 + lr];
      bf.y = Wr1[(ka + 1) * D_ + n0 + lr];
      acc = wmma4(a, bf, acc);
    }
    const float bb = br1[n0 + lr];
#pragma unroll
    for (int j = 0; j < 8; ++j)
      hid[j + 8 * half][n0 + lr] = fmaxf(acc[j] + bb, 0.f);
  }
  __syncthreads();

  // rel = hidden @ Wr2 + br2
  for (int nt = 0; nt < 4; ++nt) {
    const int n0 = nt * 16;
    v8f acc = {};
    for (int k = 0; k < D_; k += 4) {
      const int ka = k + 2 * half;
      v2f a;
      a.x = hid[lr][ka];
      a.y = hid[lr][ka + 1];
      v2f bf;
      bf.x = Wr2[ka * D_ + n0 + lr];
      bf.y = Wr2[(ka + 1) * D_ + n0 + lr];
      acc = wmma4(a, bf, acc);
    }
    const float bb = br2[n0 + lr];
#pragma unroll
    for (int j = 0; j < 8; ++j) {
      const int row = m0 + j + 8 * half;
      const float val = acc[j] + bb;
      rel[row * D_ + n0 + lr] = val;   // x_obs (fixed)
      xv[row * D_ + n0 + lr]  = val;   // PC state init
    }
  }
}

// ---------------------------------------------------------------------------
// DDL delta update + LayerNorm, one 512-wide row per block.
// ---------------------------------------------------------------------------
__global__ __launch_bounds__(256)
void delta_ln_kernel(const float* __restrict__ x, const float* __restrict__ kdir,
                     const float* __restrict__ outp, const float* __restrict__ beta,
                     const float* __restrict__ ln_g, const float* __restrict__ ln_b,
                     float* __restrict__ out) {
  __shared__ float red[256];
  const int r = blockIdx.x;
  const int t = threadIdx.x;
  const int base = r * ED_;
  const float k0 = kdir[base + t], k1 = kdir[base + t + 256];
  const float o0 = outp[base + t], o1 = outp[base + t + 256];
  const float x0 = x[base + t],    x1 = x[base + t + 256];

  red[t] = k0 * k0 + k1 * k1; __syncthreads();
  for (int o = 128; o > 0; o >>= 1) { if (t < o) red[t] += red[t + o]; __syncthreads(); }
  const float ksq = red[0]; __syncthreads();

  red[t] = k0 * o0 + k1 * o1; __syncthreads();
  for (int o = 128; o > 0; o >>= 1) { if (t < o) red[t] += red[t + o]; __syncthreads(); }
  const float kdot = red[0]; __syncthreads();

  // k = kraw/sqrt(ksq+eps); delta = x + beta * k * (k.out)
  const float factor = beta[r] * kdot / (ksq + 1e-12f);
  const float d0 = x0 + factor * k0, d1 = x1 + factor * k1;

  red[t] = d0 + d1; __syncthreads();
  for (int o = 128; o > 0; o >>= 1) { if (t < o) red[t] += red[t + o]; __syncthreads(); }
  const float mean = red[0] * (1.f / ED_); __syncthreads();

  const float dm0 = d0 - mean, dm1 = d1 - mean;
  red[t] = dm0 * dm0 + dm1 * dm1; __syncthreads();
  for (int o = 128; o > 0; o >>= 1) { if (t < o) red[t] += red[t + o]; __syncthreads(); }
  const float var = red[0] * (1.f / ED_);
  const float inv = 1.f / sqrtf(var + 1e-5f);
  out[base + t]       = dm0 * inv * ln_g[t]       + ln_b[t];
  out[base + t + 256] = dm1 * inv * ln_g[t + 256] + ln_b[t + 256];
}

// ---------------------------------------------------------------------------
// beta = 2*sigmoid(x @ Wdb + bdb); one wave per row.
// ---------------------------------------------------------------------------
__global__ __launch_bounds__(32)
void beta_kernel(const float* __restrict__ x, const float* __restrict__ Wdb,
                 const float* __restrict__ bdb, float* __restrict__ beta) {
  const int row = blockIdx.x;
  const int lane = threadIdx.x;
  const float* xr = x + row * ED_;
  float s = 0.f;
  for (int c = lane; c < ED_; c += 32) s += xr[c] * Wdb[c];
  for (int o = 16; o > 0; o >>= 1) s += __shfl_xor(s, o, 32);
  if (lane == 0) beta[row] = 2.f / (1.f + __expf(-(s + bdb[0])));
}

// ---------------------------------------------------------------------------
// PC solve machinery.  Cycle r corresponds to edge e=63+r (i>=1):
//   (C x)[r] = x[i-1] + x[63+r] - x[j-1];  row_n = 3 for all rows.
// ---------------------------------------------------------------------------
__global__ void build_edges(int* __restrict__ eit, int* __restrict__ ejt) {
  const int e = blockIdx.x * blockDim.x + threadIdx.x;
  if (e >= NE_) return;
  int i = 0, rem = e;
  while (rem >= 63 - i) { rem -= 63 - i; ++i; }
  eit[e] = i;
  ejt[e] = i + 1 + rem;
}

__global__ __launch_bounds__(256)
void colmean_kernel(const float* __restrict__ xv, float* __restrict__ mu) {
  __shared__ float red[256];
  const int d = blockIdx.x;
  float s = 0.f;
  for (int r = threadIdx.x; r < NE_; r += 256) s += xv[r * D_ + d];
  red[threadIdx.x] = s; __syncthreads();
  for (int o = 128; o > 0; o >>= 1) { if (threadIdx.x < o) red[threadIdx.x] += red[threadIdx.x + o]; __syncthreads(); }
  if (threadIdx.x == 0) mu[d] = red[0] * (1.f / NE_);
}

__global__ void cx_kernel(const float* __restrict__ xv, const int* __restrict__ eit,
                          const int* __restrict__ ejt, float* __restrict__ cx) {
  const int idx = blockIdx.x * blockDim.x + threadIdx.x;
  if (idx >= NCYC_ * D_) return;
  const int r = idx >> 6, d = idx & 63;
  const int e = 63 + r;
  const int i = eit[e], j = ejt[e];
  cx[idx] = xv[(i - 1) * D_ + d] + xv[e * D_ + d] - xv[(j - 1) * D_ + d];
}

// (C^T y)[e]:  non-tree edge -> y[e-63];  tree edge (0,t): +sum over cycles
// with i=t (contiguous r-range), -sum over cycles with j=t (scattered).
static __device__ __forceinline__ float ct_val(const float* __restrict__ cx,
                                               int e, int d) {
  if (e >= 63) return cx[(e - 63) * D_ + d];
  const int t = e + 1;
  float s = 0.f;
  const int off = 63 * t - (t * (t - 1)) / 2;   // edge offset of row i=t
  const int r0 = off - 63;
  const int cnt = 63 - t;
  for (int q = 0; q < cnt; ++q) s += cx[(r0 + q) * D_ + d];
  for (int i2 = 1; i2 < t; ++i2) {
    const int e2 = 63 * i2 - (i2 * (i2 - 1)) / 2 + (t - i2 - 1);
    s -= cx[(e2 - 63) * D_ + d];
  }
  return s;
}

__global__ void update_kernel(float* __restrict__ xv, const float* __restrict__ cx,
                              const float* __restrict__ xobs,
                              const float* __restrict__ mu, int mode) {
  const int idx = blockIdx.x * blockDim.x + threadIdx.x;
  if (idx >= NE_ * D_) return;
  const int e = idx >> 6, d = idx & 63;
  const float cty = ct_val(cx, e, d);
  const float x = xv[idx];
  if (mode == 0) {
    // gradient step: d/dx [ mean((x-obs)^2) + L_CTX*mean((Cx)^2) + L_VAR*mean(var) ]
    const float c_data = 2.f / (float)(NE_ * D_);
    const float c_ctx  = 1.0f * 2.f / (float)(NCYC_ * D_);
    const float c_var  = 0.5f * 2.f / (float)(NE_ * D_);
    const float g = c_data * (x - xobs[idx]) + c_ctx * cty + c_var * (x - mu[d]);
    xv[idx] = x - 0.1f * g;                     // PC_LR
  } else {
    xv[idx] = x - 0.8f * cty * (1.f / 3.f);     // PC_ALPHA, row_n == 3
  }
}

__global__ __launch_bounds__(256)
void violation_kernel(const float* __restrict__ cx, float* __restrict__ out) {
  __shared__ float red[256];
  float s = 0.f;
  for (int idx = threadIdx.x; idx < NCYC_ * D_; idx += 256) {
    const float v = cx[idx];
    s += v * v;
  }
  red[threadIdx.x] = s; __syncthreads();
  for (int o = 128; o > 0; o >>= 1) { if (threadIdx.x < o) red[threadIdx.x] += red[threadIdx.x + o]; __syncthreads(); }
  if (threadIdx.x == 0) out[0] = sqrtf(red[0]);
}

// ---------------------------------------------------------------------------
extern "C" void kernel_launch(void* const* d_in, const int* in_sizes, int n_in,
                              void* d_out, int out_size, void* d_ws, size_t ws_size,
                              hipStream_t stream) {
  (void)in_sizes; (void)n_in; (void)out_size; (void)ws_size;
  const float* x   = (const float*)d_in[0];
  const float* Wq  = (const float*)d_in[1];
  const float* bq  = (const float*)d_in[2];
  const float* Wk  = (const float*)d_in[3];
  const float* bk  = (const float*)d_in[4];
  const float* Wv  = (const float*)d_in[5];
  const float* bv  = (const float*)d_in[6];
  const float* Wo  = (const float*)d_in[7];
  const float* bo  = (const float*)d_in[8];
  const float* Wr1 = (const float*)d_in[9];
  const float* br1 = (const float*)d_in[10];
  const float* Wr2 = (const float*)d_in[11];
  const float* br2 = (const float*)d_in[12];
  const float* lng = (const float*)d_in[13];
  const float* lnb = (const float*)d_in[14];
  const float* Wdk = (const float*)d_in[15];
  const float* bdk = (const float*)d_in[16];
  const float* Wdb = (const float*)d_in[17];
  const float* bdb = (const float*)d_in[18];
  float* out = (float*)d_out;

  float* w    = (float*)d_ws;
  float* Q    = w;  w += ROWS_ * ED_;    // (B,H,S,D)
  float* Kt   = w;  w += ROWS_ * ED_;
  float* V    = w;  w += ROWS_ * ED_;
  float* attn = w;  w += ROWS_ * ED_;    // (B,S,H*D)
  float* outp = w;  w += ROWS_ * ED_;    // attn @ Wo + bo
  float* kdir = w;  w += ROWS_ * ED_;    // raw x @ Wdk + bdk
  float* beta = w;  w += ROWS_;
  float* rel  = w;  w += NE_ * D_;       // x_obs
  float* xv   = w;  w += NE_ * D_;       // PC state
  float* cx   = w;  w += NCYC_ * D_;
  float* mu   = w;  w += D_;
  int* eit = (int*)w;
  int* ejt = eit + NE_;

  const dim3 gg(ED_ / 16, ROWS_ / 16);
  gemm_wmma<<<gg, 32, 0, stream>>>(x, Wq, bq, Q,  ED_, ED_, 1);
  gemm_wmma<<<gg, 32, 0, stream>>>(x, Wk, bk, Kt, ED_, ED_, 1);
  gemm_wmma<<<gg, 32, 0, stream>>>(x, Wv, bv, V,  ED_, ED_, 1);
  gemm_wmma<<<gg, 32, 0, stream>>>(x, Wdk, bdk, kdir, ED_, ED_, 0);

  build_edges<<<(NE_ + 255) / 256, 256, 0, stream>>>(eit, ejt);
  beta_kernel<<<ROWS_, 32, 0, stream>>>(x, Wdb, bdb, beta);

  attn_kernel<<<B_ * H_, 128, 0, stream>>>(Q, Kt, V, attn);
  gemm_wmma<<<gg, 32, 0, stream>>>(attn, Wo, bo, outp, ED_, ED_, 0);
  delta_ln_kernel<<<ROWS_, 256, 0, stream>>>(x, kdir, outp, beta, lng, lnb, out);

  // rel[0,0] + PC solve + violation
  rel_kernel<<<NE_ / 16, 32, 0, stream>>>(Q, Kt, Wr1, br1, Wr2, br2, rel, xv);
  const int cxB = (NCYC_ * D_ + 255) / 256;
  const int upB = (NE_ * D_ + 255) / 256;
  for (int it = 0; it < 5; ++it) {
    colmean_kernel<<<D_, 256, 0, stream>>>(xv, mu);
    cx_kernel<<<cxB, 256, 0, stream>>>(xv, eit, ejt, cx);
    update_kernel<<<upB, 256, 0, stream>>>(xv, cx, rel, mu, 0);   // grad step
    cx_kernel<<<cxB, 256, 0, stream>>>(xv, eit, ejt, cx);
    update_kernel<<<upB, 256, 0, stream>>>(xv, cx, rel, mu, 1);   // projection
  }
  cx_kernel<<<cxB, 256, 0, stream>>>(xv, eit, ejt, cx);
  violation_kernel<<<1, 256, 0, stream>>>(cx, out + ROWS_ * ED_);
}